// GaussianPMIModel_69166153334974
// MI455X (gfx1250) — compile-verified
//
#include <hip/hip_runtime.h>
#include <hip/hip_bf16.h>

typedef __attribute__((ext_vector_type(2))) float v2f;
typedef __attribute__((ext_vector_type(8))) float v8f;

#define NTRAIN   50000
#define DFEAT    32
#define NEVAL    100000
#define NPAIR    496
#define OUTC     529            // 1 + 32 + 496

// cov GEMM partials: 125 waves x 400 samples = 50000 exactly -> no bounds
// checks in the inner loop (removes exec-mask branching seen in round 1 asm)
#define COV_BLOCKS 25
#define COV_WPB    5            // waves per block (160 threads)
#define COV_WAVES  (COV_BLOCKS * COV_WPB)   // 125
#define CHUNK      400          // samples per wave, multiple of 4
#define PART_STRIDE 832         // 3*256 tile floats + 64 colsum floats

// ---------------------------------------------------------------------------
// Kernel 1: per-wave partial S = X^T X (upper tiles 00,01,11) + column sums,
// using V_WMMA_F32_16X16X4_F32. Deterministic (fixed chunk ownership).
// ---------------------------------------------------------------------------
__global__ void cov_partial_kernel(const float* __restrict__ X,
                                   float* __restrict__ part) {
    const int lane = threadIdx.x & 31;
    const int wave = (blockIdx.x * blockDim.x + threadIdx.x) >> 5;
    const int c    = lane & 15;      // feature within 16-wide tile
    const int half = lane >> 4;      // 0: K=0,1   1: K=2,3
    const int sbase = wave * CHUNK;

    v8f acc00 = {}; v8f acc01 = {}; v8f acc11 = {};
    float sumlo = 0.0f, sumhi = 0.0f;

    for (int k = 0; k < CHUNK; k += 4) {
        const int s0 = sbase + k + (half << 1);   // in [sbase, sbase+CHUNK)
        const int s1 = s0 + 1;

        // unconditional, coalesced: lanes 0-15 read rows s,s+1; 16-31 s+2,s+3
        const float f0lo = X[s0 * DFEAT + c];
        const float f0hi = X[s0 * DFEAT + 16 + c];
        const float f1lo = X[s1 * DFEAT + c];
        const float f1hi = X[s1 * DFEAT + 16 + c];

        // stream-ahead hint (global_prefetch_b8); speculative, tail-safe
        __builtin_prefetch(&X[(s0 + 8) * DFEAT + c], 0, 1);

        sumlo += f0lo + f1lo;
        sumhi += f0hi + f1hi;

        v2f alo = {f0lo, f1lo};      // A frag, features [0,16)
        v2f ahi = {f0hi, f1hi};      // A frag, features [16,32)
        // D = A x B + C : S tiles (i<=j only; 10 recovered by symmetry)
        acc00 = __builtin_amdgcn_wmma_f32_16x16x4_f32(false, alo, false, alo,
                                                      (short)0, acc00, false, false);
        acc01 = __builtin_amdgcn_wmma_f32_16x16x4_f32(false, alo, false, ahi,
                                                      (short)0, acc01, false, false);
        acc11 = __builtin_amdgcn_wmma_f32_16x16x4_f32(false, ahi, false, ahi,
                                                      (short)0, acc11, false, false);
    }

    float* p = part + wave * PART_STRIDE;
    // C/D layout: VGPR r, lane l -> M = r + (l>=16 ? 8 : 0), N = l&15
    #pragma unroll
    for (int r = 0; r < 8; ++r) {
        const int m = r + (half << 3);
        p[0 * 256 + m * 16 + c] = acc00[r];
        p[1 * 256 + m * 16 + c] = acc01[r];
        p[2 * 256 + m * 16 + c] = acc11[r];
    }
    // column sums: feature f partial = [768+f] + [768+32+f]
    p[768 + half * 32 + c]      = sumlo;
    p[768 + half * 32 + 16 + c] = sumhi;
}

// S(i,j) lookup for i <= j from the three stored tiles
__device__ __forceinline__ float s_at(const float* sS, int i, int j) {
    if (j < 16)  return sS[i * 16 + j];
    if (i < 16)  return sS[256 + i * 16 + (j - 16)];
    return sS[512 + (i - 16) * 16 + (j - 16)];
}

// ---------------------------------------------------------------------------
// Kernel 2: reduce partials -> mean, inv_std, and per-pair (i,j,a,b,c0) table
// ---------------------------------------------------------------------------
__global__ void finalize_kernel(const float* __restrict__ part,
                                float* __restrict__ wmean,
                                float* __restrict__ winv,
                                int*   __restrict__ pidx,
                                float* __restrict__ pa,
                                float* __restrict__ pb,
                                float* __restrict__ pc0) {
    __shared__ float sS[768];
    __shared__ float sMean[DFEAT];
    __shared__ float sVar[DFEAT];
    const int tid = threadIdx.x;
    const float invNm1 = 1.0f / (float)(NTRAIN - 1);

    for (int t = tid; t < 768; t += blockDim.x) {
        float s = 0.0f;
        for (int b = 0; b < COV_WAVES; ++b) s += part[b * PART_STRIDE + t];
        sS[t] = s;
    }
    if (tid < DFEAT) {
        float s = 0.0f;
        for (int b = 0; b < COV_WAVES; ++b)
            s += part[b * PART_STRIDE + 768 + tid] +
                 part[b * PART_STRIDE + 800 + tid];
        const float m = s / (float)NTRAIN;
        sMean[tid] = m;
        wmean[tid] = m;
    }
    __syncthreads();
    if (tid < DFEAT) {
        const float v = (s_at(sS, tid, tid) -
                         (float)NTRAIN * sMean[tid] * sMean[tid]) * invNm1;
        sVar[tid] = v;
        winv[tid] = rsqrtf(v);
    }
    __syncthreads();
    if (tid < NPAIR) {
        // combinations order: (0,1),(0,2),...,(0,31),(1,2),...
        int i = 0, start = 0;
        while (start + (DFEAT - 1 - i) <= tid) { start += DFEAT - 1 - i; ++i; }
        const int j = i + 1 + (tid - start);
        const float cij = (s_at(sS, i, j) -
                           (float)NTRAIN * sMean[i] * sMean[j]) * invNm1;
        const float rho = cij * rsqrtf(sVar[i] * sVar[j]);
        const float om  = 1.0f - rho * rho;
        pa[tid]  = 0.5f * rho * rho / om;
        pb[tid]  = rho / om;
        pc0[tid] = 0.5f * logf(om);
        pidx[tid] = i | (j << 8);
    }
}

// ---------------------------------------------------------------------------
// Kernel 3: eval. One wave per output row; lanes = consecutive output columns
// (coalesced). Row residuals + pair table staged in LDS. NT stores: 212 MB
// output > 192 MB L2, stream it past the cache.
// ---------------------------------------------------------------------------
#define EVAL_WPB 8
__global__ void eval_kernel(const float* __restrict__ x,
                            const float* __restrict__ mean,
                            const float* __restrict__ invstd,
                            const int*   __restrict__ pidx,
                            const float* __restrict__ pa,
                            const float* __restrict__ pb,
                            const float* __restrict__ pc0,
                            float* __restrict__ out) {
    __shared__ float su[EVAL_WPB][DFEAT];
    __shared__ float s_pa[NPAIR], s_pb[NPAIR], s_pc0[NPAIR];
    __shared__ int   s_ij[NPAIR];

    const int lane = threadIdx.x & 31;
    const int wv   = threadIdx.x >> 5;
    const int row  = blockIdx.x * EVAL_WPB + wv;   // grid sized exactly

    for (int t = threadIdx.x; t < NPAIR; t += blockDim.x) {
        s_pa[t]  = pa[t];
        s_pb[t]  = pb[t];
        s_pc0[t] = pc0[t];
        s_ij[t]  = pidx[t];
    }

    const float xv = __builtin_nontemporal_load(&x[row * DFEAT + lane]);
    su[wv][lane] = (xv - mean[lane]) * invstd[lane];
    __syncthreads();

    float* orow = out + (size_t)row * OUTC;
    __builtin_nontemporal_store(1.0f, &orow[lane]);      // null + singles = 1
    if (lane == 0) __builtin_nontemporal_store(1.0f, &orow[32]);

    #pragma unroll
    for (int t = 0; t < 16; ++t) {
        const int p = t * 32 + lane;
        if (p < NPAIR) {
            const int ij = s_ij[p];
            const int i = ij & 0xff, j = ij >> 8;
            const float ui = su[wv][i];
            const float uj = su[wv][j];
            const float q = s_pc0[p] + s_pa[p] * (ui * ui + uj * uj)
                          - s_pb[p] * ui * uj;
            __builtin_nontemporal_store(__expf(q), &orow[33 + p]);
        }
    }
}

// ---------------------------------------------------------------------------
extern "C" void kernel_launch(void* const* d_in, const int* in_sizes, int n_in,
                              void* d_out, int out_size, void* d_ws, size_t ws_size,
                              hipStream_t stream) {
    (void)in_sizes; (void)n_in; (void)out_size; (void)ws_size;
    const float* X = (const float*)d_in[0];   // X_train 50000x32
    const float* x = (const float*)d_in[1];   // x 100000x32
    float* out = (float*)d_out;

    float* ws    = (float*)d_ws;
    float* part  = ws;                                   // 125*832 = 104000
    float* wmean = ws + COV_WAVES * PART_STRIDE;         // 32
    float* winv  = wmean + DFEAT;                        // 32
    float* pa    = winv + DFEAT;                         // 496
    float* pb    = pa + NPAIR;                           // 496
    float* pc0   = pb + NPAIR;                           // 496
    int*   pidx  = (int*)(pc0 + NPAIR);                  // 496

    cov_partial_kernel<<<dim3(COV_BLOCKS), dim3(COV_WPB * 32), 0, stream>>>(X, part);
    finalize_kernel<<<dim3(1), dim3(512), 0, stream>>>(part, wmean, winv,
                                                       pidx, pa, pb, pc0);
    eval_kernel<<<dim3(NEVAL / EVAL_WPB), dim3(EVAL_WPB * 32), 0, stream>>>(
        x, wmean, winv, pidx, pa, pb, pc0, out);
}